// NNCorr_21672404975756
// MI455X (gfx1250) — compile-verified
//
#include <hip/hip_runtime.h>
#include <hip/hip_bf16.h>

typedef float v2f __attribute__((ext_vector_type(2)));
typedef float v8f __attribute__((ext_vector_type(8)));
typedef unsigned long long u64;

#define B_    2
#define N1_   8192
#define N2_   8192
#define TSTRIP 16   // 16x16 tiles per wave along N2 -> 256-wide strip

// clamp to >= 0 in ONE op: signed-int max on IEEE bits (negatives -> +0.0)
__device__ __forceinline__ int clamp0_bits(float v) {
    int bi = __float_as_int(v);
    return bi > 0 ? bi : 0;          // v_max_i32
}
__device__ __forceinline__ float clamp0(float v) {
    return __int_as_float(clamp0_bits(v));
}

// ---------------- init workspace: packed (d2bits<<32)|idx minima ------------
__global__ void nn_init_ws(u64* ws, int n) {
    int i = blockIdx.x * blockDim.x + threadIdx.x;
    if (i < n) ws[i] = ~0ull;
}

// ---------------- main fused kernel ----------------------------------------
// One wave computes a 16(row) x 256(col) strip of corr_mat for one batch.
// d2 tile via V_WMMA_F32_16X16X4_F32 with homogeneous-coordinate trick.
// Argmin reductions run on raw d2 (sqrt is monotonic) so the min dataflow
// never waits on v_sqrt_f32.
__global__ void __launch_bounds__(256)
nn_corr_kernel(const float* __restrict__ x1, const float* __restrict__ x2,
               float* __restrict__ corr,
               u64* __restrict__ wsrow,   // [B*N1] row minima (argmin over n)
               u64* __restrict__ wscol)   // [B*N2] col minima (argmin over m)
{
    const int mtiles  = N1_ / 16;            // 512
    const int nstrips = N2_ / (16 * TSTRIP); // 32

    int wave = (int)((blockIdx.x * blockDim.x + threadIdx.x) >> 5);
    int lane = threadIdx.x & 31;
    int half = lane >> 4;        // 0: lanes 0-15, 1: lanes 16-31
    int l16  = lane & 15;

    int b  = wave / (mtiles * nstrips);
    int r  = wave % (mtiles * nstrips);
    int mt = r / nstrips;
    int ns = r % nstrips;
    int m0    = mt * 16;
    int nbase = ns * 16 * TSTRIP;

    // ---- A matrix (16x4 f32): lane L holds M = L%16.
    // half 0 supplies K=0,1 (ax, ay); half 1 supplies K=2,3 (az, ||a||^2).
    const float* ap = x1 + ((size_t)b * N1_ + m0 + l16) * 3;
    float ax = ap[0], ay = ap[1], az = ap[2];
    float asq = ax * ax + ay * ay + az * az;
    v2f A;
    A.x = half ? az  : ax;
    A.y = half ? asq : ay;

    // running row minima on d2 (index = column n); j => row m0 + j + 8*half
    float rv[8];
    int   rn[8];
#pragma unroll
    for (int j = 0; j < 8; ++j) { rv[j] = 3.4e38f; rn[j] = 0; }

    float* crow = corr + ((size_t)b * N1_ + m0) * N2_;

#pragma unroll 2
    for (int t = 0; t < TSTRIP; ++t) {
        int n0 = nbase + t * 16;

        // ---- B matrix (4x16 f32): lane L holds N = L%16.
        // half 0 supplies K=0,1 (-2bx, -2by); half 1 supplies K=2,3 (-2bz, 1).
        const float* bp = x2 + ((size_t)b * N2_ + n0 + l16) * 3;
        float bx = bp[0], by = bp[1], bz = bp[2];
        float bsq = bx * bx + by * by + bz * bz;
        v2f Bm;
        Bm.x = half ? (-2.0f * bz) : (-2.0f * bx);
        Bm.y = half ? 1.0f         : (-2.0f * by);

        // ---- C matrix: ||b_n||^2 broadcast down each column (N = lane%16)
        v8f c;
#pragma unroll
        for (int j = 0; j < 8; ++j) c[j] = bsq;

        // d2[m][n] = ||a||^2 + ||b||^2 - 2 a.b
        v8f d = __builtin_amdgcn_wmma_f32_16x16x4_f32(
            false, A, false, Bm, (short)0, c, false, false);

        // ---- store path: v_max_i32 clamp + raw v_sqrt_f32, NT b32 clause
#pragma unroll
        for (int j = 0; j < 8; ++j) {
            float v = __builtin_amdgcn_sqrtf(clamp0(d[j]));
            __builtin_nontemporal_store(
                v, &crow[(size_t)(j + half * 8) * N2_ + n0 + l16]);
        }

        // ---- argmin paths on raw d2 (independent of sqrt results)
#pragma unroll
        for (int j = 0; j < 8; ++j) {
            if (d[j] < rv[j]) { rv[j] = d[j]; rn[j] = n0 + l16; }
        }

        // column minima (argmin over m) for this tile's 16 columns
        float cmv = d[0];
        int   cmi = m0 + half * 8;
#pragma unroll
        for (int j = 1; j < 8; ++j)
            if (d[j] < cmv) { cmv = d[j]; cmi = m0 + j + half * 8; }
        // combine the two 8-row halves (lane L and L^16 share column n0+l16)
        float ov = __shfl_xor(cmv, 16, 32);
        int   oi = __shfl_xor(cmi, 16, 32);
        if (ov < cmv || (ov == cmv && oi < cmi)) { cmv = ov; cmi = oi; }
        if (half == 0) {
            // clamp once at flush so packed bits are monotonic (non-negative)
            u64 pk = ((u64)(unsigned)clamp0_bits(cmv) << 32) | (unsigned)cmi;
            atomicMin(&wscol[(size_t)b * N2_ + n0 + l16], pk);
        }
    }

    // ---- row minima: reduce across the 16 lanes of each half
#pragma unroll
    for (int j = 0; j < 8; ++j) {
        float v = rv[j];
        int   n = rn[j];
#pragma unroll
        for (int s = 1; s < 16; s <<= 1) {
            float ov = __shfl_xor(v, s, 32);
            int   on = __shfl_xor(n, s, 32);
            if (ov < v || (ov == v && on < n)) { v = ov; n = on; }
        }
        if (l16 == 0) {
            u64 pk = ((u64)(unsigned)clamp0_bits(v) << 32) | (unsigned)n;
            atomicMin(&wsrow[(size_t)b * N1_ + m0 + j + half * 8], pk);
        }
    }
}

// ---------------- unpack indices to float outputs ---------------------------
__global__ void nn_finalize(const u64* __restrict__ wsrow,
                            const u64* __restrict__ wscol,
                            float* __restrict__ idx21,  // [B*N1]
                            float* __restrict__ idx12)  // [B*N2]
{
    int i = blockIdx.x * blockDim.x + threadIdx.x;
    if (i < B_ * N1_) idx21[i] = (float)(unsigned)(wsrow[i] & 0xFFFFFFFFull);
    if (i < B_ * N2_) idx12[i] = (float)(unsigned)(wscol[i] & 0xFFFFFFFFull);
}

extern "C" void kernel_launch(void* const* d_in, const int* in_sizes, int n_in,
                              void* d_out, int out_size, void* d_ws, size_t ws_size,
                              hipStream_t stream) {
    const float* x1 = (const float*)d_in[0];
    const float* x2 = (const float*)d_in[1];
    float* out = (float*)d_out;

    const size_t szX1   = (size_t)B_ * N1_ * 3;           // 49152
    const size_t szX2   = (size_t)B_ * N2_ * 3;           // 49152
    const size_t szCorr = (size_t)B_ * N1_ * N2_;         // 134217728

    float* out_x1    = out;
    float* out_x2    = out + szX1;
    float* out_corr  = out + szX1 + szX2;
    float* out_idx12 = out + szX1 + szX2 + szCorr;                  // [B,N2]
    float* out_idx21 = out + szX1 + szX2 + szCorr + (size_t)B_*N2_; // [B,N1]

    u64* wsrow = (u64*)d_ws;              // B*N1 packed minima
    u64* wscol = wsrow + (size_t)B_ * N1_;// B*N2 packed minima
    const int nws = B_ * N1_ + B_ * N2_;  // 32768 u64 = 256 KB

    // pass-through inputs
    hipMemcpyAsync(out_x1, x1, szX1 * sizeof(float), hipMemcpyDeviceToDevice, stream);
    hipMemcpyAsync(out_x2, x2, szX2 * sizeof(float), hipMemcpyDeviceToDevice, stream);

    nn_init_ws<<<(nws + 255) / 256, 256, 0, stream>>>(wsrow, nws);

    // waves = B * (N1/16) * (N2/256) = 32768 ; 8 waves/block -> 4096 blocks
    const int mtiles  = N1_ / 16;
    const int nstrips = N2_ / (16 * TSTRIP);
    const int totalWaves = B_ * mtiles * nstrips;
    const int blocks = totalWaves / 8;
    nn_corr_kernel<<<blocks, 256, 0, stream>>>(x1, x2, out_corr, wsrow, wscol);

    nn_finalize<<<(B_ * N2_ + 255) / 256, 256, 0, stream>>>(wsrow, wscol,
                                                            out_idx21, out_idx12);
}